// CenterLossLayer_66786741453069
// MI455X (gfx1250) — compile-verified
//
#include <hip/hip_runtime.h>
#include <hip/hip_bf16.h>

typedef __attribute__((ext_vector_type(2))) float v2f;
typedef __attribute__((ext_vector_type(8))) float v8f;

constexpr int BATCH = 16384;
constexpr int FEAT  = 512;
constexpr int NCLS  = 1024;
constexpr float ALPHA = 0.5f;

// ---------------------------------------------------------------------------
// Kernel 1: zero the scatter workspace (delta[NCLS*FEAT] ++ counts[NCLS]).
// Must run every call: harness poisons ws once and never re-poisons.
// ---------------------------------------------------------------------------
__global__ void cl_zero_ws(float* __restrict__ ws, int n) {
    int i = blockIdx.x * blockDim.x + threadIdx.x;
    if (i < n) ws[i] = 0.0f;
}

// ---------------------------------------------------------------------------
// Kernel 2: per-sample distance via WMMA diag(A * A^T) + scatter of residuals.
// One wave32 owns a tile of 16 consecutive samples.
//   lane L: m = L%16 (sample in tile), half = L/16 selects K-subcolumns.
// f32 16x4 A-matrix layout (ISA 7.12.2): lane m holds K = 2*half + {0,1}
// in VGPR0/VGPR1. The 4x16 B layout is its transpose, so A's registers are
// reused directly as B:  acc += A * A^T, diagonal = sum_k diff^2.
// ---------------------------------------------------------------------------
__global__ __launch_bounds__(256) void cl_main(
        const float* __restrict__ feat,     // [BATCH, FEAT]
        const float* __restrict__ cent,     // [NCLS, FEAT]
        const int*   __restrict__ labels,   // [BATCH]
        float*       __restrict__ out_dist, // [BATCH]
        float*       __restrict__ delta,    // [NCLS, FEAT] (ws, zeroed)
        float*       __restrict__ counts)   // [NCLS]       (ws, zeroed)
{
    const int gid  = blockIdx.x * blockDim.x + threadIdx.x;
    const int wave = gid >> 5;
    const int lane = gid & 31;
    const int m    = lane & 15;
    const int half = lane >> 4;          // 0: K={0,1}, 1: K={2,3} of each chunk
    const int row0 = wave * 16;
    const int row  = row0 + m;

    const int lbl = labels[row];
    const float* frow = feat + (size_t)row * FEAT;
    const float* crow = cent + (size_t)lbl * FEAT;
    float*       drow = delta + (size_t)lbl * FEAT;

    v8f acc = {};   // 16x16 f32 accumulator, only the diagonal is consumed

    #pragma unroll 4
    for (int k = 0; k < FEAT; k += 4) {
        const int col = k + half * 2;
        v2f f = *(const v2f*)(frow + col);   // global_load_b64
        v2f c = *(const v2f*)(crow + col);   // global_load_b64 (L2-resident)
        v2f d;
        d.x = c.x - f.x;                     // diff = xc - features
        d.y = c.y - f.y;

        // per-class residual scatter (delta_centers = onehot^T * diff)
        atomicAdd(drow + col + 0, d.x);      // global_atomic_add_f32
        atomicAdd(drow + col + 1, d.y);

        // acc += A * A^T  (A == B registers by transpose-symmetry of layouts)
        acc = __builtin_amdgcn_wmma_f32_16x16x4_f32(
                /*neg_a=*/false, d, /*neg_b=*/false, d,
                /*c_mod=*/(short)0, acc, /*reuse_a=*/false, /*reuse_b=*/false);
    }

    // Diagonal (m,m) of the 16x16 f32 C/D layout:
    //   m in 0..7  -> lane m,      VGPR m
    //   m in 8..15 -> lane m+8,   VGPR m-8
    int vidx = -1, srow = 0;
    if (lane < 8)                    { vidx = lane;      srow = row0 + lane;     }
    else if (lane >= 16 && lane < 24){ vidx = lane - 16; srow = row0 + lane - 8; }
    if (vidx >= 0) {
        float dist = acc[0];
        #pragma unroll
        for (int i = 1; i < 8; ++i)
            if (vidx == i) dist = acc[i];    // cndmask select chain
        out_dist[srow] = dist;
    }

    // per-class sample count (one lane per sample)
    if (lane < 16) atomicAdd(&counts[lbl], 1.0f);
}

// ---------------------------------------------------------------------------
// Kernel 3: new_centers = centers - ALPHA * delta / (counts + 1)
// One block per class row, float4-vectorized (128 threads * 4 = 512 feats).
// ---------------------------------------------------------------------------
__global__ __launch_bounds__(128) void cl_finalize(
        const float* __restrict__ cent,
        const float* __restrict__ delta,
        const float* __restrict__ counts,
        float*       __restrict__ out_centers)
{
    const int c = blockIdx.x;
    const int t = threadIdx.x;               // 0..127
    const float s = ALPHA / (counts[c] + 1.0f);

    const float4* cv = (const float4*)(cent  + (size_t)c * FEAT);
    const float4* dv = (const float4*)(delta + (size_t)c * FEAT);
    float4*       ov = (float4*)(out_centers + (size_t)c * FEAT);

    float4 cc = cv[t];
    float4 dd = dv[t];
    float4 r;
    r.x = cc.x - s * dd.x;
    r.y = cc.y - s * dd.y;
    r.z = cc.z - s * dd.z;
    r.w = cc.w - s * dd.w;
    ov[t] = r;
}

// ---------------------------------------------------------------------------
extern "C" void kernel_launch(void* const* d_in, const int* in_sizes, int n_in,
                              void* d_out, int out_size, void* d_ws, size_t ws_size,
                              hipStream_t stream) {
    const float* feat   = (const float*)d_in[0];   // [16384, 512] f32
    const float* cent   = (const float*)d_in[1];   // [1024, 512] f32
    const int*   labels = (const int*)  d_in[2];   // [16384] i32

    float* out_dist    = (float*)d_out;            // [16384, 1]
    float* out_centers = (float*)d_out + BATCH;    // [1024, 512]

    float* delta  = (float*)d_ws;                  // [1024*512]
    float* counts = delta + (size_t)NCLS * FEAT;   // [1024]

    const int nz = NCLS * FEAT + NCLS;
    cl_zero_ws<<<(nz + 255) / 256, 256, 0, stream>>>(delta, nz);

    // 16384 samples / 16 per wave = 1024 waves = 32768 threads
    const int threads = (BATCH / 16) * 32;
    cl_main<<<threads / 256, 256, 0, stream>>>(feat, cent, labels,
                                               out_dist, delta, counts);

    cl_finalize<<<NCLS, 128, 0, stream>>>(cent, delta, counts, out_centers);
}